// GCNEncoder_80307298500865
// MI455X (gfx1250) — compile-verified
//
#include <hip/hip_runtime.h>

typedef __attribute__((ext_vector_type(2))) float v2f;
typedef __attribute__((ext_vector_type(8))) float v8f;

#define N_NODES 100000
#define N_EDGES 1600000
#define IN_C    10
#define HID_C   50
#define OUT_C   32

// padded strides
#define K1P   12    // IN_C padded to mult of 4
#define N1P   64    // HID_C padded to mult of 16 (hs1 stride)
#define LDA1  52    // acc1 / h2 stride: HID_C padded to mult of 4 (K of GEMM2)
#define K2P   52
#define N2P   32

// workspace layout (float offsets)
#define DEG_OFF   0                          // 100,000 (deg -> dinv in place)
#define W1P_OFF   100352                     // 12*64   = 768
#define W2P_OFF   101504                     // 52*32   = 1,664
#define XP1_OFF   103424                     // 100,000*12 = 1,200,000
#define HS1_OFF   (XP1_OFF + 1200000)        // 100,000*64 = 6,400,000 (later hs2, stride 32)
#define ACC1_OFF  (HS1_OFF + 6400000)        // 100,000*52 = 5,200,000 (later h2, in place)
#define ACC2_OFF  (ACC1_OFF + 5200000)       // 100,000*32 = 3,200,000
// total = 16,103,424 floats ~= 64.4 MB

// ---------------------------------------------------------------- utilities
__global__ void fill_kernel(float* __restrict__ p, float v, int n) {
    int i = blockIdx.x * blockDim.x + threadIdx.x;
    if (i < n) p[i] = v;
}

__global__ void deg_kernel(const long long* __restrict__ dst, float* __restrict__ deg, int n) {
    int e = blockIdx.x * blockDim.x + threadIdx.x;
    if (e < n) atomicAdd(&deg[(int)dst[e]], 1.0f);
}

__global__ void rsqrt_kernel(float* __restrict__ deg, int n) {
    int i = blockIdx.x * blockDim.x + threadIdx.x;
    if (i < n) deg[i] = rsqrtf(deg[i]);   // deg >= 1 always (self-loop init)
}

// zero-pad a [R, CS] row-major matrix into [RP, CP] (zeros outside R x CS)
__global__ void pad2d_kernel(const float* __restrict__ src, float* __restrict__ dstp,
                             int R, int CS, int RP, int CP) {
    int i = blockIdx.x * blockDim.x + threadIdx.x;
    if (i < RP * CP) {
        int r = i / CP, c = i - r * CP;
        dstp[i] = (r < R && c < CS) ? src[r * CS + c] : 0.0f;
    }
}

// ------------------------------------------------- WMMA f32 GEMM, dinv-scaled
// Y[m, n] = (sum_k Xp[m,k] * Wp[k,n]) * dinv[m]
// Xp: [N_NODES, KP] (zero-padded), Wp: [KP, NT*16] (zero-padded),
// Y: [N_NODES, NT*16]. Branch-free inner loop; one wave does MT 16x16 tiles,
// keeping all B fragments in registers.
template<int KP, int NT, int MT>
__global__ void gemm_wmma_scale(const float* __restrict__ Xp, const float* __restrict__ Wp,
                                const float* __restrict__ dinv, float* __restrict__ Y) {
    constexpr int NP     = NT * 16;
    constexpr int KSTEPS = KP / 4;
    constexpr int MG     = (N_NODES / 16) / MT;   // m-tile groups (6250/MT)
    const int wave = (int)((blockIdx.x * blockDim.x + threadIdx.x) >> 5);
    const int lane = threadIdx.x & 31;
    if (wave >= MG * NT) return;                  // wave-uniform exit

    const int mg    = wave / NT;
    const int nt    = wave % NT;
    const int khalf = lane >> 4;                  // 0: lanes 0-15, 1: lanes 16-31
    const int l15   = lane & 15;
    const int col   = nt * 16 + l15;

    // preload all B fragments (loop-invariant across M)
    v2f b[KSTEPS];
    #pragma unroll
    for (int s = 0; s < KSTEPS; ++s) {
        const int ka = s * 4 + khalf * 2;
        b[s].x = Wp[(ka    ) * NP + col];
        b[s].y = Wp[(ka + 1) * NP + col];
    }

    #pragma unroll
    for (int t = 0; t < MT; ++t) {
        const int mt  = mg * MT + t;
        const int row = mt * 16 + l15;
        v8f c = {};
        #pragma unroll
        for (int s = 0; s < KSTEPS; ++s) {
            const int ka = s * 4 + khalf * 2;
            v2f a;
            a.x = Xp[row * KP + ka];
            a.y = Xp[row * KP + ka + 1];
            c = __builtin_amdgcn_wmma_f32_16x16x4_f32(
                    /*neg_a=*/false, a, /*neg_b=*/false, b[s],
                    /*c_mod=*/(short)0, c, /*reuse_a=*/false, /*reuse_b=*/false);
        }
        #pragma unroll
        for (int v = 0; v < 8; ++v) {
            const int m = mt * 16 + v + khalf * 8;
            Y[m * NP + col] = c[v] * dinv[m];     // padded cols are exact zeros
        }
    }
}

// ------------------------------------------------------------ edge scatter
// acc[dst, c] += hs[src, c], c < C — one wave per edge, lanes sweep channels.
template<int C, int LDH, int LDA>
__global__ void scatter_kernel(const long long* __restrict__ src,
                               const long long* __restrict__ dst,
                               const float* __restrict__ hs,
                               float* __restrict__ acc) {
    const int wave = (int)((blockIdx.x * blockDim.x + threadIdx.x) >> 5);
    const int lane = threadIdx.x & 31;
    if (wave >= N_EDGES) return;
    __builtin_prefetch(src + wave + 2048, 0, 0);  // speculative; dropped if OOB
    __builtin_prefetch(dst + wave + 2048, 0, 0);
    const int s = (int)src[wave];
    const int d = (int)dst[wave];
    for (int c = lane; c < C; c += 32)
        atomicAdd(&acc[d * LDA + c], hs[s * LDH + c]);
}

// --------------------------------------- combine: out = dinv*(acc+hs)+b (+relu)
template<int C, int LDA, int LDH, int LDO, bool RELU>
__global__ void combine_kernel(const float* __restrict__ acc, const float* __restrict__ hs,
                               const float* __restrict__ dinv, const float* __restrict__ bias,
                               float* __restrict__ out) {
    int i = blockIdx.x * blockDim.x + threadIdx.x;
    if (i < N_NODES * C) {
        const int node = i / C;
        const int ch   = i - node * C;
        float v = dinv[node] * (acc[node * LDA + ch] + hs[node * LDH + ch]) + bias[ch];
        if (RELU) v = v > 0.0f ? v : 0.0f;
        out[node * LDO + ch] = v;
    }
}

// ---------------------------------------------------------------------------
extern "C" void kernel_launch(void* const* d_in, const int* in_sizes, int n_in,
                              void* d_out, int out_size, void* d_ws, size_t ws_size,
                              hipStream_t stream) {
    const float*     x   = (const float*)d_in[0];
    const long long* ei  = (const long long*)d_in[1];   // int64 [2, E]
    const float*     W1  = (const float*)d_in[2];
    const float*     b1  = (const float*)d_in[3];
    const float*     W2  = (const float*)d_in[4];
    const float*     b2  = (const float*)d_in[5];
    float*           out = (float*)d_out;

    float* ws   = (float*)d_ws;
    float* deg  = ws + DEG_OFF;    // becomes dinv
    float* W1p  = ws + W1P_OFF;
    float* W2p  = ws + W2P_OFF;
    float* Xp1  = ws + XP1_OFF;
    float* hs1  = ws + HS1_OFF;    // later reused as hs2 (stride 32)
    float* acc1 = ws + ACC1_OFF;   // stride 52; later h2 (in place, K-padded)
    float* acc2 = ws + ACC2_OFF;

    const long long* srcv = ei;
    const long long* dstv = ei + N_EDGES;

    const int T = 256;
    auto blks = [](long long n, int t) { return (int)((n + t - 1) / t); };

    // ---- init (every call: deterministic; harness does not re-zero ws)
    fill_kernel<<<blks(N_NODES, T), T, 0, stream>>>(deg, 1.0f, N_NODES);          // self-loop
    fill_kernel<<<blks(N_NODES * LDA1, T), T, 0, stream>>>(acc1, 0.0f, N_NODES * LDA1);
    fill_kernel<<<blks(N_NODES * OUT_C, T), T, 0, stream>>>(acc2, 0.0f, N_NODES * OUT_C);

    // ---- zero-padded operands for branch-free WMMA loops
    pad2d_kernel<<<blks(N_NODES * K1P, T), T, 0, stream>>>(x,  Xp1, N_NODES, IN_C,  N_NODES, K1P);
    pad2d_kernel<<<blks(K1P * N1P, T), T, 0, stream>>>(W1, W1p, IN_C,  HID_C, K1P, N1P);
    pad2d_kernel<<<blks(K2P * N2P, T), T, 0, stream>>>(W2, W2p, HID_C, OUT_C, K2P, N2P);

    // ---- degree -> dinv
    deg_kernel<<<blks(N_EDGES, T), T, 0, stream>>>(dstv, deg, N_EDGES);
    rsqrt_kernel<<<blks(N_NODES, T), T, 0, stream>>>(deg, N_NODES);

    // ---- layer 1: hs1 = (x @ W1) * dinv  (K=12 padded, 4 col tiles, stride 64)
    {
        long long waves = (long long)((N_NODES / 16) / 5) * 4;   // MT=5
        gemm_wmma_scale<K1P, 4, 5><<<blks(waves * 32, T), T, 0, stream>>>(Xp1, W1p, deg, hs1);
    }
    scatter_kernel<HID_C, N1P, LDA1><<<blks((long long)N_EDGES * 32, T), T, 0, stream>>>(
        srcv, dstv, hs1, acc1);
    // h2 = relu(dinv*(acc1 + hs1) + b1), in place over acc1 (stays K-padded: cols 50,51 = 0)
    combine_kernel<HID_C, LDA1, N1P, LDA1, true><<<blks((long long)N_NODES * HID_C, T), T, 0, stream>>>(
        acc1, hs1, deg, b1, acc1);

    // ---- layer 2: hs2 = (h2 @ W2) * dinv  (K=52 padded, 2 col tiles, stride 32; reuse hs1 buffer)
    {
        long long waves = (long long)((N_NODES / 16) / 5) * 2;   // MT=5
        gemm_wmma_scale<K2P, 2, 5><<<blks(waves * 32, T), T, 0, stream>>>(acc1, W2p, deg, hs1);
    }
    scatter_kernel<OUT_C, N2P, N2P><<<blks((long long)N_EDGES * 32, T), T, 0, stream>>>(
        srcv, dstv, hs1, acc2);
    // z = dinv*(acc2 + hs2) + b2  -> d_out
    combine_kernel<OUT_C, N2P, N2P, N2P, false><<<blks((long long)N_NODES * OUT_C, T), T, 0, stream>>>(
        acc2, hs1, deg, b2, out);
}